// ASRCTCLoss_35046933135628
// MI455X (gfx1250) — compile-verified
//
#include <hip/hip_runtime.h>
#include <math.h>

// CTC loss (PyTorch semantics, reduction='mean', zero_infinity) for
// B=32, T=1024 (sliced to 1023), V=1000, L=256 (sliced to 255), S=511.
//
// CDNA5-specific paths used:
//  - global_load_async_to_lds_b128 (ASYNCcnt) to stage logits rows / emit rows
//    into LDS, overlapped with compute (s_wait_asynccnt).
//  - wave32 shuffles (warpSize==32) for the softmax reductions.
// No WMMA: the op has no matmul structure (banded log-semiring scan +
// HBM-bound softmax), so the matrix pipe cannot help.

#define NEGV (-1e30f)

// ---------- CDNA5 async global->LDS helpers (probe-guarded) ----------
typedef int v4i __attribute__((vector_size(16)));
typedef __attribute__((address_space(1))) v4i gv4i;  // global (HIP "__device__")
typedef __attribute__((address_space(3))) v4i lv4i;  // LDS (HIP "__shared__")

__device__ __forceinline__ void async_ld_b128(const void* g, void* l) {
#if __has_builtin(__builtin_amdgcn_global_load_async_to_lds_b128)
  // signature: (v4i AS1*, v4i AS3*, imm offset, imm cpol)
  __builtin_amdgcn_global_load_async_to_lds_b128(
      (gv4i*)(size_t)g, (lv4i*)(unsigned)(size_t)l, 0, 0);
#else
  // generic LDS pointer: low 32 bits == LDS byte address
  unsigned lds_addr = (unsigned)(size_t)l;
  asm volatile("global_load_async_to_lds_b128 %0, %1, off"
               :: "v"(lds_addr), "v"(g) : "memory");
#endif
}

__device__ __forceinline__ void wait_async0() {
#if __has_builtin(__builtin_amdgcn_s_wait_asynccnt)
  __builtin_amdgcn_s_wait_asynccnt(0);
#else
  asm volatile("s_wait_asynccnt 0" ::: "memory");
#endif
}

// ---------- Kernel A: target lengths + extended label sequence ----------
// tlen[b] = sum(mask[b]) - 1 ; ext[b][s] = (s odd) ? targets[b][1 + s/2] : 0
__global__ __launch_bounds__(512) void ctc_prep(
    const int* __restrict__ targets, const unsigned char* __restrict__ mask,
    int* __restrict__ ext, int* __restrict__ tlen) {
  const int b = blockIdx.x, tid = threadIdx.x;
  __shared__ int cnt;
  if (tid == 0) cnt = 0;
  __syncthreads();
  int v = (tid < 256 && mask[b * 256 + tid]) ? 1 : 0;
  for (int o = 16; o; o >>= 1) v += __shfl_down(v, o, 32);
  if ((tid & 31) == 0) atomicAdd(&cnt, v);
  __syncthreads();
  if (tid == 0) tlen[b] = cnt - 1;
  int e = 0;
  if ((tid & 1) && tid < 511) e = targets[b * 256 + 1 + (tid >> 1)];
  ext[b * 512 + tid] = e;
}

// ---------- Kernel B: per-row logsumexp + emit gather ----------
// emit[b][t][s] = (s < 2*tlen+1) ? logits[b][t][ext[s]] - lse : NEG
__global__ __launch_bounds__(256) void ctc_lse_emit(
    const float* __restrict__ logits, const int* __restrict__ ext,
    const int* __restrict__ tlen, float* __restrict__ emit) {
  const int t = blockIdx.x;   // 0..1022
  const int b = blockIdx.y;   // 0..31
  const int tid = threadIdx.x;
  const float* row = logits + ((size_t)b * 1024 + t) * 1000;

  __shared__ __align__(16) float srow[1024];  // 4 KB row staging (+24 slack)
  __shared__ float red[8];

  // Async-stage the row: 256 lanes x 16 B = 4096 B (24 floats of safe slack:
  // row t<=1022 so the next row of the same tensor backs the over-read).
  async_ld_b128(row + tid * 4, &srow[tid * 4]);
  wait_async0();
  __syncthreads();

  // block max over V=1000 (wave32 shuffle + LDS combine)
  float m = -3.402823466e+38f;
  for (int i = tid; i < 1000; i += 256) m = fmaxf(m, srow[i]);
  for (int o = 16; o; o >>= 1) m = fmaxf(m, __shfl_xor(m, o, 32));
  if ((tid & 31) == 0) red[tid >> 5] = m;
  __syncthreads();
  m = red[0];
  for (int w = 1; w < 8; ++w) m = fmaxf(m, red[w]);

  // block sum of exp(x - m)
  float s = 0.f;
  for (int i = tid; i < 1000; i += 256) s += __expf(srow[i] - m);
  for (int o = 16; o; o >>= 1) s += __shfl_xor(s, o, 32);
  __syncthreads();                      // all reads of red[] done
  if ((tid & 31) == 0) red[tid >> 5] = s;
  __syncthreads();
  s = 0.f;
  for (int w = 0; w < 8; ++w) s += red[w];
  const float lse = m + __logf(s);

  const int smax = 2 * tlen[b] + 1;     // <= 511
  float* erow = emit + ((size_t)b * 1023 + t) * 512;
  const int* eb = ext + b * 512;
  for (int sIdx = tid; sIdx < 512; sIdx += 256) {
    float val = NEGV;
    if (sIdx < 511 && sIdx < smax) val = srow[eb[sIdx]] - lse;
    erow[sIdx] = val;
  }
}

// ---------- Kernel C: alpha recursion (one workgroup per batch) ----------
__global__ __launch_bounds__(512) void ctc_scan(
    const float* __restrict__ emit, const int* __restrict__ ext,
    const int* __restrict__ tlen, float* __restrict__ loss_b) {
  const int b = blockIdx.x;
  const int tid = threadIdx.x;

  __shared__ float alpha[2][514];               // [s+2] layout, guards at 0,1
  __shared__ __align__(16) float ebuf[2][512];  // double-buffered emit rows

  const int* eb = ext + b * 512;
  const float* base = emit + (size_t)b * 1023 * 512;

  const int myext = (tid < 511) ? eb[tid] : 0;
  const bool allow =
      (tid >= 2) && (tid < 511) && (myext != 0) && (myext != eb[tid - 2]);

  // t = 0 init (emit row 0 is already validity-masked)
  if (tid < 2) { alpha[0][tid] = NEGV; alpha[1][tid] = NEGV; }
  float a0 = NEGV;
  if (tid == 0) a0 = base[0];
  else if (tid == 1) a0 = base[1];
  alpha[0][tid + 2] = a0;

  // preload emit row 1 into ebuf[1] (128 lanes x 16 B = 2048 B)
  if (tid < 128) async_ld_b128(base + 512 + tid * 4, &ebuf[1][tid * 4]);
  wait_async0();
  __syncthreads();

  for (int t = 1; t < 1023; ++t) {
    const int cur = t & 1, prv = cur ^ 1;
    // prefetch next row into the buffer we just finished reading
    if (t + 1 < 1023 && tid < 128)
      async_ld_b128(base + (size_t)(t + 1) * 512 + tid * 4, &ebuf[prv][tid * 4]);
    if (tid < 511) {
      const float a1 = alpha[prv][tid + 2];
      const float a2 = alpha[prv][tid + 1];
      const float a3 = allow ? alpha[prv][tid] : NEGV;
      const float mm = fmaxf(a1, fmaxf(a2, a3));
      const float nw = mm +
          __logf(__expf(a1 - mm) + __expf(a2 - mm) + __expf(a3 - mm)) +
          ebuf[cur][tid];
      alpha[cur][tid + 2] = nw;
    }
    wait_async0();      // prefetch landed (overlapped with the LSE above)
    __syncthreads();    // publish alpha[cur] + ebuf[prv] to all 16 waves
  }

  if (tid == 0) {
    const int tl = tlen[b];
    const int fb = 1022 & 1;  // final buffer == 0
    const int i1 = 2 * tl;
    const float l1 = alpha[fb][i1 + 2];
    const float l2 = (tl > 0) ? alpha[fb][i1 + 1] : NEGV;
    const float mm = fmaxf(l1, l2);
    float loss = -(mm + __logf(__expf(l1 - mm) + __expf(l2 - mm)));
    if (loss > 1e20f) loss = 0.f;
    const float den = (float)(tl > 1 ? tl : 1);
    loss_b[b] = loss / den;
  }
}

// ---------- Kernel D: deterministic mean ----------
__global__ void ctc_reduce(const float* __restrict__ loss_b,
                           float* __restrict__ out) {
  if (threadIdx.x == 0 && blockIdx.x == 0) {
    float s = 0.f;
    for (int b = 0; b < 32; ++b) s += loss_b[b];
    out[0] = s / 32.0f;
  }
}

extern "C" void kernel_launch(void* const* d_in, const int* in_sizes, int n_in,
                              void* d_out, int out_size, void* d_ws, size_t ws_size,
                              hipStream_t stream) {
  (void)in_sizes; (void)n_in; (void)out_size; (void)ws_size;
  const float* logits = (const float*)d_in[0];               // (32,1024,1000) f32
  const int* targets = (const int*)d_in[1];                  // (32,256) i32
  const unsigned char* mask = (const unsigned char*)d_in[2]; // (32,256) bool

  char* ws = (char*)d_ws;
  int*   ext    = (int*)(ws + 0);            // 32*512*4  = 64 KiB
  int*   tlen   = (int*)(ws + 65536);        // 128 B
  float* loss_b = (float*)(ws + 65536 + 256);
  float* emit   = (float*)(ws + 131072);     // 32*1023*512*4 ~= 64 MiB

  ctc_prep<<<dim3(32), dim3(512), 0, stream>>>(targets, mask, ext, tlen);
  ctc_lse_emit<<<dim3(1023, 32), dim3(256), 0, stream>>>(logits, ext, tlen, emit);
  ctc_scan<<<dim3(32), dim3(512), 0, stream>>>(emit, ext, tlen, loss_b);
  ctc_reduce<<<dim3(1), dim3(32), 0, stream>>>(loss_b, (float*)d_out);
}